// CosineSimilarityLoss_33251636806170
// MI455X (gfx1250) — compile-verified
//
#include <hip/hip_runtime.h>
#include <math.h>

#ifndef __has_builtin
#define __has_builtin(x) 0
#endif

#define NT 256
#define LOG2E 1.4426950408889634f

typedef int v4i __attribute__((ext_vector_type(4)));

// ---- CDNA5 async global->LDS copy (16B per lane), ASYNCcnt-tracked ----
__device__ __forceinline__ void async_copy16(const float4* gsrc, float4* lds_dst) {
#if __has_builtin(__builtin_amdgcn_global_load_async_to_lds_b128)
  __builtin_amdgcn_global_load_async_to_lds_b128((v4i*)gsrc, (v4i*)lds_dst, 0, 0);
#else
  // Fallback: synchronous copy through VGPRs (still correct; no ASYNCcnt).
  *lds_dst = *gsrc;
#endif
}

template <int N>
__device__ __forceinline__ void wait_async() {
#if __has_builtin(__builtin_amdgcn_s_wait_asynccnt)
  __builtin_amdgcn_s_wait_asynccnt(N);
#else
  asm volatile("s_wait_asynccnt %0" ::"i"(N) : "memory");
#endif
  asm volatile("" ::: "memory");  // compiler barrier before LDS read
}

__device__ __forceinline__ float fast_exp(float x) {
#if __has_builtin(__builtin_amdgcn_exp2f)
  return __builtin_amdgcn_exp2f(x * LOG2E);
#else
  return exp2f(x * LOG2E);
#endif
}

// One workgroup per row: max, exp-sums, per-row masked cosine.
__global__ __launch_bounds__(NT) void cosloss_rows_kernel(
    const float* __restrict__ logits, const long long* __restrict__ targets,
    float* __restrict__ ws_cos, float* __restrict__ ws_msk, int V) {
  const int row = blockIdx.x;
  const int tid = threadIdx.x;
  const float* rp = logits + (size_t)row * (size_t)V;
  const float4* rp4 = (const float4*)rp;
  const int nv4 = V >> 2;
  const int nfull = nv4 / NT;  // full async-staged chunks for Pass B

  __shared__ float4 stage[2][NT];  // async staging, double-buffered (8 KB)
  __shared__ float red0[NT];
  __shared__ float red1[NT];

  // Issue Pass-B chunk 0 into LDS *now*: the async DMA engine fills stage[0]
  // while Pass A streams the row from HBM (overlap instead of serialize).
  if (nfull > 0) async_copy16(rp4 + tid, &stage[0][tid]);

  // ---------- Pass A: row max (HBM streaming, with prefetch) ----------
  float m = -INFINITY;
  for (int i = tid; i < nv4; i += NT) {
    __builtin_prefetch(rp4 + i + 4 * NT, 0, 0);  // global_prefetch_b8 (speculative)
    float4 v = rp4[i];
    m = fmaxf(fmaxf(fmaxf(v.x, v.y), fmaxf(v.z, v.w)), m);
  }
  red0[tid] = m;
  __syncthreads();
  for (int s = NT / 2; s > 0; s >>= 1) {
    if (tid < s) red0[tid] = fmaxf(red0[tid], red0[tid + s]);
    __syncthreads();
  }
  const float rowmax = red0[0];
  __syncthreads();

  // ---------- Pass B: sum(e), sum(e^2); row is L2-resident now ----------
  // Async double-buffered global->LDS pipeline; each thread stages and
  // consumes its own 16B slot, so no workgroup barrier is needed inside.
  float s1 = 0.0f, s2 = 0.0f;
  for (int c = 0; c < nfull; ++c) {
    if (c + 1 < nfull) {
      async_copy16(rp4 + (size_t)(c + 1) * NT + tid, &stage[(c + 1) & 1][tid]);
      wait_async<1>();  // chunk c finished; chunk c+1 in flight
    } else {
      wait_async<0>();
    }
    float4 v = stage[c & 1][tid];
    float e0 = fast_exp(v.x - rowmax);
    float e1 = fast_exp(v.y - rowmax);
    float e2 = fast_exp(v.z - rowmax);
    float e3 = fast_exp(v.w - rowmax);
    s1 += (e0 + e1) + (e2 + e3);
    s2 = fmaf(e0, e0, s2);
    s2 = fmaf(e1, e1, s2);
    s2 = fmaf(e2, e2, s2);
    s2 = fmaf(e3, e3, s2);
  }
  // Tail (nv4 % NT vec4 elements): direct loads
  for (int i = nfull * NT + tid; i < nv4; i += NT) {
    float4 v = rp4[i];
    float e0 = fast_exp(v.x - rowmax);
    float e1 = fast_exp(v.y - rowmax);
    float e2 = fast_exp(v.z - rowmax);
    float e3 = fast_exp(v.w - rowmax);
    s1 += (e0 + e1) + (e2 + e3);
    s2 = fmaf(e0, e0, s2);
    s2 = fmaf(e1, e1, s2);
    s2 = fmaf(e2, e2, s2);
    s2 = fmaf(e3, e3, s2);
  }

  red0[tid] = s1;
  red1[tid] = s2;
  __syncthreads();
  for (int s = NT / 2; s > 0; s >>= 1) {
    if (tid < s) {
      red0[tid] += red0[tid + s];
      red1[tid] += red1[tid + s];
    }
    __syncthreads();
  }

  if (tid == 0) {
    float S1 = red0[0], S2 = red1[0];
    long long t = targets[row];
    float lt = rp[t];
    float et = fast_exp(lt - rowmax);
    // cos = (e_t/S1) / max(sqrt(S2)/S1, eps)   (one-hot norm clamp = 1)
    float pt = et / S1;
    float nrm = sqrtf(S2) / S1;
    float cosv = pt / fmaxf(nrm, 1e-8f);
    float msk = (t != 0) ? 1.0f : 0.0f;
    ws_cos[row] = cosv * msk;
    ws_msk[row] = msk;
  }
}

// Deterministic final reduction over rows: out = 1 - sum(cos)/(sum(mask)+eps)
__global__ __launch_bounds__(NT) void cosloss_final_kernel(
    const float* __restrict__ ws_cos, const float* __restrict__ ws_msk,
    float* __restrict__ out, int R) {
  __shared__ float a[NT];
  __shared__ float b[NT];
  const int tid = threadIdx.x;
  float sc = 0.0f, sm = 0.0f;
  for (int i = tid; i < R; i += NT) {
    sc += ws_cos[i];
    sm += ws_msk[i];
  }
  a[tid] = sc;
  b[tid] = sm;
  __syncthreads();
  for (int s = NT / 2; s > 0; s >>= 1) {
    if (tid < s) {
      a[tid] += a[tid + s];
      b[tid] += b[tid + s];
    }
    __syncthreads();
  }
  if (tid == 0) out[0] = 1.0f - a[0] / (b[0] + 1e-8f);
}

extern "C" void kernel_launch(void* const* d_in, const int* in_sizes, int n_in,
                              void* d_out, int out_size, void* d_ws, size_t ws_size,
                              hipStream_t stream) {
  const float* logits = (const float*)d_in[0];
  const long long* targets = (const long long*)d_in[1];
  const int R = in_sizes[1];              // B*S rows (4096)
  const int V = in_sizes[0] / R;          // vocab (32000)
  float* ws_cos = (float*)d_ws;           // R floats
  float* ws_msk = ws_cos + R;             // R floats

  cosloss_rows_kernel<<<R, NT, 0, stream>>>(logits, targets, ws_cos, ws_msk, V);
  cosloss_final_kernel<<<1, NT, 0, stream>>>(ws_cos, ws_msk, (float*)d_out, R);
}